// ABELSpline_86431921864802
// MI455X (gfx1250) — compile-verified
//
#include <hip/hip_runtime.h>
#include <hip/hip_bf16.h>
#include <math.h>

typedef __attribute__((ext_vector_type(2))) float v2f;
typedef __attribute__((ext_vector_type(8))) float v8f;

#define INPUT_DIM    128
#define DENSITY      259
#define OUTPUT_DIM   64
#define INDIRECT_DIM 512
#define BATCH        1024
#define NTILES       36          // 4 direct + 32 indirect 16-column tiles
#define TILES_PER_WAVE 9

__global__ __launch_bounds__(256) void ABELSpline_kernel(
    const float* __restrict__ x,      // [1024,128]
    const float* __restrict__ dtab,   // [33152,64]
    const float* __restrict__ itab,   // [33152,512]
    float* __restrict__ out)          // [1024,64]
{
  __shared__ float sBuf[2][INPUT_DIM * 4];                 // spline weights
  __shared__ int   iBuf[2][INPUT_DIM * 4];                 // gather rows
  __shared__ float accB[2][OUTPUT_DIM + INDIRECT_DIM];     // 576 accumulators

  const int tid    = threadIdx.x;
  const int lane   = tid & 31;
  const int waveId = tid >> 5;      // 0..7
  const int wv     = waveId & 3;    // sub-wave within a batch row (owns 9 tiles)
  const int bl     = waveId >> 2;   // local batch row 0/1
  const int b      = blockIdx.x * 2 + bl;

  // ---------------- Phase 1: spline weights + gather indices ----------------
  {
    const int   d  = wv * 32 + lane;          // each wave covers 32 dims
    const float xv = x[b * INPUT_DIM + d];
    const float t  = xv * 256.0f;             // (DENSITY-3) scale
    const float fl = floorf(t);
    const float fr = t - fl;                  // frac = mod(t,1)
    const int   i0 = (int)fl;
    const float om = 1.0f - fr;
    const float fr2 = fr * fr, fr3 = fr2 * fr;
    float4 sv;                                // cubic B-spline basis (bias 3,2,1,0)
    sv.x = om * om * om * (1.0f / 6.0f);
    sv.y = (3.0f * fr3 - 6.0f * fr2 + 4.0f) * (1.0f / 6.0f);
    sv.z = (-3.0f * fr3 + 3.0f * fr2 + 3.0f * fr + 1.0f) * (1.0f / 6.0f);
    sv.w = fr3 * (1.0f / 6.0f);
    int4 iv;                                  // global table rows
    iv.x = d * DENSITY + ((i0 + 0) % DENSITY);
    iv.y = d * DENSITY + ((i0 + 1) % DENSITY);
    iv.z = d * DENSITY + ((i0 + 2) % DENSITY);
    iv.w = d * DENSITY + ((i0 + 3) % DENSITY);
    *(float4*)&sBuf[bl][d * 4] = sv;
    *(int4*)  &iBuf[bl][d * 4] = iv;
  }
  __syncthreads();

  // ---------------- Phase 2: gather + WMMA f32 16x16x4 accumulation ----------
  // A[16,4]: lanes 0-15 hold (M=lane, K=0/1), lanes 16-31 hold (M=lane-16, K=2/3)
  // B[4,16]: weights replicated over N; VGPR0 = rows K0|K2, VGPR1 = rows K1|K3
  const bool hi        = lane >= 16;
  const int  colInTile = lane & 15;

  const float* baseJ[TILES_PER_WAVE];
  int          shiftJ[TILES_PER_WAVE];   // log2(row stride in elements)
  #pragma unroll
  for (int j = 0; j < TILES_PER_WAVE; ++j) {
    const int g = wv + j * 4;            // global tile 0..35
    if (g < 4) { baseJ[j] = dtab + g * 16 + colInTile;        shiftJ[j] = 6; }
    else       { baseJ[j] = itab + (g - 4) * 16 + colInTile;  shiftJ[j] = 9; }
  }

  const v8f vzero = {0.f, 0.f, 0.f, 0.f, 0.f, 0.f, 0.f, 0.f};
  v8f c[TILES_PER_WAVE];
  #pragma unroll
  for (int j = 0; j < TILES_PER_WAVE; ++j) c[j] = vzero;

  for (int d = 0; d < INPUT_DIM; ++d) {
    const float4 sv = *(const float4*)&sBuf[bl][d * 4];  // broadcast LDS read
    const int4   iv = *(const int4*)  &iBuf[bl][d * 4];
    v2f bmat;
    bmat.x = hi ? sv.z : sv.x;                           // B rows K0 / K2
    bmat.y = hi ? sv.w : sv.y;                           // B rows K1 / K3
    const long rA = hi ? (long)iv.z : (long)iv.x;
    const long rB = hi ? (long)iv.w : (long)iv.y;
    #pragma unroll
    for (int j = 0; j < TILES_PER_WAVE; ++j) {
      v2f a;
      a.x = baseJ[j][rA << shiftJ[j]];                   // (M=col, K=0|2)
      a.y = baseJ[j][rB << shiftJ[j]];                   // (M=col, K=1|3)
      c[j] = __builtin_amdgcn_wmma_f32_16x16x4_f32(
          /*neg_a=*/false, a, /*neg_b=*/false, bmat,
          /*c_mod=*/(short)0, c[j], /*reuse_a=*/false, /*reuse_b=*/false);
    }
  }

  // C columns are replicated over N; lane 0 holds M=0..7, lane 16 holds M=8..15.
  #pragma unroll
  for (int j = 0; j < TILES_PER_WAVE; ++j) {
    const int g  = wv + j * 4;
    const int o0 = (g < 4) ? g * 16 : (OUTPUT_DIM + (g - 4) * 16);
    if (lane == 0) {
      #pragma unroll
      for (int r = 0; r < 8; ++r) accB[bl][o0 + r] = c[j][r];
    } else if (lane == 16) {
      #pragma unroll
      for (int r = 0; r < 8; ++r) accB[bl][o0 + 8 + r] = c[j][r];
    }
  }
  __syncthreads();

  // ---------------- Phase 3: anti-symmetric exponential + store --------------
  if (tid < 128) {
    const int bl3 = tid >> 6;
    const int o   = tid & 63;
    const int bg  = blockIdx.x * 2 + bl3;
    const float  direct = accB[bl3][o];
    const float* z = &accB[bl3][OUTPUT_DIM + o * 8];     // [2][4] exp terms
    const float eb1 = -1.3862943611198906f;              // -2 ln 2
    const float eb2 = -2.1972245773362196f;              // -2 ln 3
    const float eb3 = -2.7725887222397811f;              // -2 ln 4
    const float pos = __expf(z[0]) + __expf(z[1] + eb1) +
                      __expf(z[2] + eb2) + __expf(z[3] + eb3);
    const float neg = __expf(z[4]) + __expf(z[5] + eb1) +
                      __expf(z[6] + eb2) + __expf(z[7] + eb3);
    out[bg * OUTPUT_DIM + o] = direct + pos - neg;
  }
}

extern "C" void kernel_launch(void* const* d_in, const int* in_sizes, int n_in,
                              void* d_out, int out_size, void* d_ws, size_t ws_size,
                              hipStream_t stream) {
  const float* x    = (const float*)d_in[0];   // [1024*128]
  const float* dtab = (const float*)d_in[1];   // [33152*64]
  const float* itab = (const float*)d_in[2];   // [33152*512]
  float* out = (float*)d_out;                  // [1024*64]
  ABELSpline_kernel<<<BATCH / 2, 256, 0, stream>>>(x, dtab, itab, out);
}